// RoiAlign_47845935677665
// MI455X (gfx1250) — compile-verified
//
#include <hip/hip_runtime.h>
#include <math.h>

// CDNA5 / gfx1250: wave32, WMMA F32 16x16x4.
typedef __attribute__((ext_vector_type(2))) float v2f;
typedef __attribute__((ext_vector_type(8))) float v8f;

#define POOL 7
#define CDIM 256
#define CT 64            // channel tile processed per LDS pass
#define RSTRIDE 80       // padded LDS row stride (floats): bank-offset 16 per row slot

// Sparse x-interp weight matrix: Ax[m][k] nonzero only at k==2m (1-lx) and k==2m+1 (lx).
__device__ __forceinline__ float ax_weight(int m, int k, float lxv) {
    if (m >= POOL) return 0.0f;
    if (k == 2 * m)     return 1.0f - lxv;
    if (k == 2 * m + 1) return lxv;
    return 0.0f;
}

__global__ __launch_bounds__(256) void RoiAlign_47845935677665_kernel(
    const float* __restrict__ boxes,  // [B,N,4] (y1,x1,y2,x2)
    const float* __restrict__ meta,   // [B,12], meta[4]=H_img, meta[5]=W_img
    const float* __restrict__ p2,     // [B,256,256,256]
    const float* __restrict__ p3,     // [B,128,128,256]
    const float* __restrict__ p4,     // [B, 64, 64,256]
    const float* __restrict__ p5,     // [B, 32, 32,256]
    float* __restrict__ out,          // [B,N,7,7,256]
    int B, int N)
{
    // y-lerped strip: R[py][slot j (14 used, 2 zero-pad)][channel tile]
    __shared__ float R[POOL][16][RSTRIDE];   // 35,840 bytes

    const int boxIdx = blockIdx.x;
    const int b = boxIdx / N;

    const float y1 = boxes[(size_t)boxIdx * 4 + 0];
    const float x1 = boxes[(size_t)boxIdx * 4 + 1];
    const float y2 = boxes[(size_t)boxIdx * 4 + 2];
    const float x2 = boxes[(size_t)boxIdx * 4 + 3];
    const float bh = y2 - y1, bw = x2 - x1;

    // roi_level = clip(4 + round(log2(sqrt(w*h)*sqrt(area)/224)), 2, 5)
    const float imgH = meta[4], imgW = meta[5];
    const float lvlF = log2f(sqrtf(fmaxf(bw * bh, 1e-12f)) * sqrtf(imgH * imgW) * (1.0f / 224.0f));
    const int level = (int)fminf(fmaxf(4.0f + rintf(lvlF), 2.0f), 5.0f);

    const float* fp;
    int H;
    switch (level) {
        case 2:  fp = p2; H = 256; break;
        case 3:  fp = p3; H = 128; break;
        case 4:  fp = p4; H = 64;  break;
        default: fp = p5; H = 32;  break;
    }
    const int W = H;
    const float hm1 = (float)(H - 1), wm1 = (float)(W - 1);
    fp += (size_t)b * H * W * CDIM;

    const int tid  = threadIdx.x;
    const int lane = tid & 31;        // wave32
    const int wid  = tid >> 5;        // 8 waves / block

    float* outBox = out + (size_t)boxIdx * (POOL * POOL * CDIM);

    // Per-lane x-lerp fraction for px = lane&15 (used to build WMMA A operand).
    const int m = lane & 15;
    float lxv = 0.0f;
    if (m < POOL) {
        const float fx  = (float)m * (1.0f / 6.0f);
        const float xsv = x1 * wm1 + fx * (x2 - x1) * wm1;
        lxv = xsv - floorf(xsv);
    }
    const int koff = (lane < 16) ? 0 : 2;   // A/B VGPR K-layout split (ISA 7.12.2)

    for (int ct = 0; ct < CDIM / CT; ++ct) {
        const int c0 = ct * CT;

        // ---------- Phase A: gather corners + y-lerp into LDS (coalesced b128) ----
        {
            const int ch = (tid & 15) * 4;    // float4 within the 64-ch tile
            const int p0 = tid >> 4;          // 16 (py,j) pairs in flight
            for (int p = p0; p < POOL * 16; p += 16) {
                const int py = p >> 4;
                const int j  = p & 15;
                float4 v = make_float4(0.f, 0.f, 0.f, 0.f);   // zero-pad slots 14,15
                if (j < 14) {
                    const int px = j >> 1, ix = j & 1;
                    const float fy  = (float)py * (1.0f / 6.0f);
                    const float ysv = y1 * hm1 + fy * (y2 - y1) * hm1;
                    const float y0f = floorf(ysv);
                    const float ly  = ysv - y0f;
                    const int   yb  = (int)y0f;
                    const int y0i = min(max(yb, 0), H - 1);
                    const int y1i = min(max(yb + 1, 0), H - 1);

                    const float fx  = (float)px * (1.0f / 6.0f);
                    const float xsv = x1 * wm1 + fx * (x2 - x1) * wm1;
                    const int   xb  = (int)floorf(xsv);
                    const int   xi  = min(max(xb + ix, 0), W - 1);

                    const float4 f0 = *(const float4*)(fp + ((size_t)y0i * W + xi) * CDIM + c0 + ch);
                    const float4 f1 = *(const float4*)(fp + ((size_t)y1i * W + xi) * CDIM + c0 + ch);
                    const float oy = 1.0f - ly;
                    v.x = oy * f0.x + ly * f1.x;
                    v.y = oy * f0.y + ly * f1.y;
                    v.z = oy * f0.z + ly * f1.z;
                    v.w = oy * f0.w + ly * f1.w;
                }
                *(float4*)&R[py][j][ch] = v;
            }
        }
        __syncthreads();

        // ---------- Phase B: x-lerp as 4 chained V_WMMA_F32_16X16X4_F32 ----------
        // D[px, c] = sum_j Ax[px][j] * R[py][j][c], K=14 padded to 16.
        for (int t = wid; t < POOL * (CT / 16); t += 8) {
            const int py = t >> 2;
            const int nt = t & 3;
            const int cc = nt * 16 + m;       // B operand N index = lane&15

            v8f acc = {};
            #pragma unroll
            for (int q = 0; q < 4; ++q) {
                const int k0 = 4 * q + koff;
                v2f a, bv;
                a.x  = ax_weight(m, k0,     lxv);
                a.y  = ax_weight(m, k0 + 1, lxv);
                bv.x = R[py][k0][cc];
                bv.y = R[py][k0 + 1][cc];
                // 8 args: (neg_a, A, neg_b, B, c_mod, C, reuse_a, reuse_b)
                acc = __builtin_amdgcn_wmma_f32_16x16x4_f32(
                    false, a, false, bv, (short)0, acc, false, false);
            }
            // D rows M=0..6 live in VGPRs 0..6, lanes 0-15 (N = lane).
            // Output is write-once streaming data: store non-temporal so the
            // 100 MB of results don't evict feature-map lines from L2 (192 MB),
            // which the corner gathers reuse heavily.
            if (lane < 16) {
                float* o = outBox + (size_t)(py * POOL) * CDIM + c0 + nt * 16 + lane;
                #pragma unroll
                for (int v = 0; v < POOL; ++v)
                    __builtin_nontemporal_store(acc[v], o + (size_t)v * CDIM);
            }
        }
        __syncthreads();
    }
}

extern "C" void kernel_launch(void* const* d_in, const int* in_sizes, int n_in,
                              void* d_out, int out_size, void* d_ws, size_t ws_size,
                              hipStream_t stream) {
    const float* boxes = (const float*)d_in[0];
    const float* meta  = (const float*)d_in[1];
    const float* p2    = (const float*)d_in[2];
    const float* p3    = (const float*)d_in[3];
    const float* p4    = (const float*)d_in[4];
    const float* p5    = (const float*)d_in[5];

    const int B = in_sizes[1] / 12;            // image_meta is [B,12]
    const int N = (in_sizes[0] / 4) / B;       // boxes is [B,N,4]

    dim3 grid(B * N), block(256);
    RoiAlign_47845935677665_kernel<<<grid, block, 0, stream>>>(
        boxes, meta, p2, p3, p4, p5, (float*)d_out, B, N);
}